// WaveletEncoder_53910429499944
// MI455X (gfx1250) — compile-verified
//
#include <hip/hip_runtime.h>
#include <stdint.h>

// 3-level 3D Haar DWT, fp32. Bandwidth-bound: ~170MB in + ~194MB out + ~9MB
// aaa re-read (L2-resident) => ~16us floor at 23.3 TB/s. Data path per block:
//   global_load_async_to_lds_b128 (ASYNCcnt) -> s_wait_asynccnt -> barrier
//   -> ds_load_b128 -> VALU butterfly -> b64 stores (NT for detail subbands).
// WMMA intentionally unused: 2-tap separable filter at 0.5 flop/byte needs
// ~12 GFLOP/s at this bandwidth; matrix ops would add layout shuffles only.

#define HAAR_K 0.35355339059327373f   // (1/sqrt(2))^3

typedef float f32x2 __attribute__((ext_vector_type(2)));
typedef float f32x4 __attribute__((ext_vector_type(4)));

// One block: (batch b, z-pair zh, two y-pairs). Each lane: two x-cells.
// blockDim = Wh, grid = 2 * Dh * (Hh/2).
template <int Dh, int Hh, int Wh>
__global__ void __launch_bounds__(Wh)
haar3d_level(const float* __restrict__ in, size_t inBatch,
             float* __restrict__ out)
{
    constexpr int W2      = 2 * Wh;               // input floats per row
    constexpr int inRow   = 2 * Wh;
    constexpr int inSlice = (2 * Hh) * (2 * Wh);
    constexpr int XT      = Wh / 2;               // lanes per y-pair

    // 2 y-pairs x 4 rows (2z x 2y) x W2 floats
    __shared__ __align__(16) float tile[2][4][W2];

    const int tid = threadIdx.x;                  // [0, Wh)
    const int xi  = tid % XT;                     // handles cells 2*xi, 2*xi+1
    const int yi  = tid / XT;                     // which y-pair (0/1)

    int bid = blockIdx.x;
    const int yb = bid % (Hh / 2);  bid /= (Hh / 2);
    const int zh = bid % Dh;        bid /= Dh;
    const int b  = bid;                           // batch in [0, 2)
    const int yh = 2 * yb + yi;

    const float* base = in + (size_t)b * inBatch
                           + (size_t)(2 * zh) * inSlice
                           + (size_t)(2 * yh) * inRow
                           + (size_t)(4 * xi);

    // LDS byte offset of this lane's 16B slot in its y-pair's row 0.
    uint32_t lbase = (uint32_t)(uintptr_t)(&tile[yi][0][0])
                   + (uint32_t)(16 * xi);

    // Async-copy 4 rows (2z x 2y) into LDS, 16B per lane per row; the XT
    // lanes of a y-pair cover each 8*Wh-byte row segment contiguously.
    #pragma unroll
    for (int r = 0; r < 4; ++r) {                 // r = 2*zoff + yoff
        const float* g = base + (size_t)(r >> 1) * inSlice
                              + (size_t)(r & 1) * inRow;
        uint32_t l = lbase + (uint32_t)(r * W2 * 4);
        asm volatile("global_load_async_to_lds_b128 %0, %1, off"
                     :: "v"(l), "v"(g) : "memory");
    }
#if __has_builtin(__builtin_amdgcn_s_wait_asynccnt)
    __builtin_amdgcn_s_wait_asynccnt(0);
#else
    asm volatile("s_wait_asynccnt 0" ::: "memory");
#endif
    __syncthreads();

    // Two 2x2x2 cells per lane: rows are (z0y0, z0y1, z1y0, z1y1).
    const f32x4 v0 = *(const f32x4*)&tile[yi][0][4 * xi];
    const f32x4 v1 = *(const f32x4*)&tile[yi][1][4 * xi];
    const f32x4 v2 = *(const f32x4*)&tile[yi][2][4 * xi];
    const f32x4 v3 = *(const f32x4*)&tile[yi][3][4 * xi];

    f32x2 r[8];
    #pragma unroll
    for (int c = 0; c < 2; ++c) {                 // cell within the 16B slot
        const int e0 = 2 * c, e1 = 2 * c + 1;
        // x pass: a = x0+x1, d = x0-x1
        const float ax0 = v0[e0] + v0[e1], dx0 = v0[e0] - v0[e1];
        const float ax1 = v1[e0] + v1[e1], dx1 = v1[e0] - v1[e1];
        const float ax2 = v2[e0] + v2[e1], dx2 = v2[e0] - v2[e1];
        const float ax3 = v3[e0] + v3[e1], dx3 = v3[e0] - v3[e1];
        // y pass (first letter = y a/d, second = x a/d), per z
        const float aa0 = ax0 + ax1, da0 = ax0 - ax1;
        const float ad0 = dx0 + dx1, dd0 = dx0 - dx1;
        const float aa1 = ax2 + ax3, da1 = ax2 - ax3;
        const float ad1 = dx2 + dx3, dd1 = dx2 - dx3;
        // z pass; subband s = 4*zd + 2*yd + xd
        r[0][c] = (aa0 + aa1) * HAAR_K;           // aaa
        r[1][c] = (ad0 + ad1) * HAAR_K;           // aad
        r[2][c] = (da0 + da1) * HAAR_K;           // ada
        r[3][c] = (dd0 + dd1) * HAAR_K;           // add
        r[4][c] = (aa0 - aa1) * HAAR_K;           // daa
        r[5][c] = (ad0 - ad1) * HAAR_K;           // dad
        r[6][c] = (da0 - da1) * HAAR_K;           // dda
        r[7][c] = (dd0 - dd1) * HAAR_K;           // ddd
    }

    const size_t N = (size_t)Dh * Hh * Wh;
    float* ob = out + (size_t)b * 8 * N
                    + ((size_t)zh * Hh + yh) * (size_t)Wh + 2 * xi;

    *(f32x2*)ob = r[0];                           // aaa: RT, re-read next level
    #pragma unroll
    for (int s = 1; s < 8; ++s)                   // details: stream past L2
        __builtin_nontemporal_store(r[s], (f32x2*)(ob + (size_t)s * N));
}

extern "C" void kernel_launch(void* const* d_in, const int* in_sizes, int n_in,
                              void* d_out, int out_size, void* d_ws, size_t ws_size,
                              hipStream_t stream) {
    (void)in_sizes; (void)n_in; (void)out_size; (void)d_ws; (void)ws_size;

    const size_t L1 = 2ull * 8 * 96 * 96 * 96;   // 14,155,776
    const size_t L2 = 2ull * 8 * 48 * 48 * 48;   //  1,769,472
    const size_t L3 = 2ull * 8 * 24 * 24 * 24;   //    221,184

    float* out = (float*)d_out;

    for (int v = 0; v < 3; ++v) {
        const float* in = (const float*)d_in[v];     // (2,1,192,192,192)
        float* o = out + (size_t)v * (L1 + L2 + L3);

        // Level 1: (2,192^3) -> (2,8,96^3); batch stride 192^3 (channel=1).
        haar3d_level<96, 96, 96><<<dim3(2 * 96 * 48), 96, 0, stream>>>(
            in, (size_t)192 * 192 * 192, o);

        // Level 2: input = level-1 subband 0 (batch stride 8*96^3).
        haar3d_level<48, 48, 48><<<dim3(2 * 48 * 24), 48, 0, stream>>>(
            o, (size_t)8 * 96 * 96 * 96, o + L1);

        // Level 3: input = level-2 subband 0 (batch stride 8*48^3).
        haar3d_level<24, 24, 24><<<dim3(2 * 24 * 12), 24, 0, stream>>>(
            o + L1, (size_t)8 * 48 * 48 * 48, o + L1 + L2);
    }
}